// Rocket2D_7748121002209
// MI455X (gfx1250) — compile-verified
//
#include <hip/hip_runtime.h>
#include <hip/hip_bf16.h>

// Rocket2D forward for MI455X (gfx1250, wave32).
// Stage 1: 3x3 conv (10 ch) as fp32 WMMA GEMM (V_WMMA_F32_16X16X4_F32).
//          A = im2col patches from a zero-padded LDS plane (branchless ds_load),
//          B = weights (lane-resident, hoisted), D cols = channels so each lane
//          stores 8 consecutive pixels with two b128 stores under one mask.
// Stage 2: 200 dilated convs + PPV; plane DMA'd to LDS with
//          GLOBAL_LOAD_ASYNC_TO_LDS_B128 (ASYNCcnt), one block per (channel,image) plane.

typedef __attribute__((ext_vector_type(2))) float v2f;
typedef __attribute__((ext_vector_type(8))) float v8f;

#define HW 28
#define NPIX 784
#define BATCH 512
#define N1 10
#define N2 200
#define PPV_T 1e-7f

// ---------------- Stage 1: layer1 conv as WMMA GEMM ----------------
__global__ __launch_bounds__(256) void rocket_layer1_wmma(
    const float* __restrict__ x, const float* __restrict__ w1,
    const float* __restrict__ b1, float* __restrict__ inter)
{
  __shared__ float xpad[901];                 // 30*30 padded plane + zero slot
  const int n   = blockIdx.x;                 // image 0..511
  const int tid = threadIdx.x;
  const float* __restrict__ xi = x + n * NPIX;

  for (int idx = tid; idx < 901; idx += 256) {
    int rr = idx / 30 - 1, cc = idx % 30 - 1;
    float v = 0.0f;
    if (idx < 900 && (unsigned)rr < (unsigned)HW && (unsigned)cc < (unsigned)HW)
      v = xi[rr * HW + cc];
    xpad[idx] = v;
  }
  __syncthreads();

  const int lane = tid & 31;
  const int wv   = tid >> 5;                  // 0..7
  const int half = lane >> 4;                 // which K-pair this lane carries
  const int lm   = lane & 15;                 // A-row (pixel) / B,D-col (channel)

  // Hoist per-lane B operand (weights) and A tap offsets out of the tile loop.
  // B[k, ch]: lane's col = lm (channel), carries taps k = s*4 + 2*half (+1).
  v2f bw[3];
  int off0[3], off1[3];                       // padded-plane offsets, 900 = zero slot
#pragma unroll
  for (int s = 0; s < 3; ++s) {
    const int g0 = s * 4 + half * 2;
    const int g1 = g0 + 1;
    bw[s].x = (lm < N1 && g0 < 9) ? w1[lm * 9 + g0] : 0.0f;
    bw[s].y = (lm < N1 && g1 < 9) ? w1[lm * 9 + g1] : 0.0f;
    off0[s] = (g0 < 9) ? ((g0 / 3) * 30 + (g0 % 3)) : -1;
    off1[s] = (g1 < 9) ? ((g1 / 3) * 30 + (g1 % 3)) : -1;
  }
  const float bias = (lm < N1) ? b1[lm] : 0.0f;
  float* const dst0 = inter + (lm * BATCH + n) * NPIX + 8 * half;

  for (int tile = wv; tile < 49; tile += 8) {
    const int p    = tile * 16 + lm;          // A-row pixel 0..783
    const int base = (p / HW) * 30 + (p % HW);// top-left of padded patch
    v8f acc = {};
#pragma unroll
    for (int s = 0; s < 3; ++s) {
      v2f a;                                  // A = im2col patch (pixel lm)
      const int i0 = (off0[s] >= 0) ? (base + off0[s]) : 900;
      const int i1 = (off1[s] >= 0) ? (base + off1[s]) : 900;
      a.x = xpad[i0];
      a.y = xpad[i1];
      acc = __builtin_amdgcn_wmma_f32_16x16x4_f32(
          /*neg_a=*/false, a, /*neg_b=*/false, bw[s],
          /*c_mod=*/(short)0, acc, /*reuse_a=*/false, /*reuse_b=*/false);
    }
    // D: col = lm (channel), rows = pixels tile*16 + 8*half + v (consecutive).
    if (lm < N1) {
      float4 lo, hi;
      lo.x = acc[0] + bias; lo.y = acc[1] + bias;
      lo.z = acc[2] + bias; lo.w = acc[3] + bias;
      hi.x = acc[4] + bias; hi.y = acc[5] + bias;
      hi.z = acc[6] + bias; hi.w = acc[7] + bias;
      float4* dst = (float4*)(dst0 + tile * 16);
      dst[0] = lo;
      dst[1] = hi;
    }
  }
}

// ---------------- Stage 2: dilated convs + PPV ----------------
// One block per (channel,image) plane (5120 blocks). Plane is DMA'd into LDS
// with async global->LDS b128 copies (ASYNCcnt), then the 8 waves sweep the
// 200 configs (25 each); lanes split output pixels; wave32 shfl_xor reduce.
__global__ __launch_bounds__(256) void rocket_layer2_ppv(
    const float* __restrict__ inter, const float* __restrict__ w2,
    const float* __restrict__ b2, const int* __restrict__ cfg,
    float* __restrict__ out)
{
  __shared__ float plane[NPIX];
  const int img = blockIdx.x;                 // img = i*512 + b
  const int tid = threadIdx.x;

  // 196 lanes x 16B async DMA: global -> LDS, tracked by ASYNCcnt.
  if (tid < NPIX / 4) {
    unsigned lds_off = (unsigned)(unsigned long long)(const void*)&plane[tid * 4];
    unsigned long long gaddr =
        (unsigned long long)(const void*)(inter + img * NPIX + tid * 4);
    asm volatile("global_load_async_to_lds_b128 %0, %1, off"
                 :: "v"(lds_off), "v"(gaddr) : "memory");
  }
  asm volatile("s_wait_asynccnt 0x0" ::: "memory");
  __syncthreads();

  const int lane = tid & 31;
  const int wv   = tid >> 5;
  const int i    = img >> 9;                  // /512
  const int bb   = img & 511;
  const float inv2t = 0.5f / PPV_T;

  for (int j = wv; j < N2; j += 8) {
    const int kr = cfg[j * 6 + 0], kc = cfg[j * 6 + 1];
    const int dr = cfg[j * 6 + 2], dc = cfg[j * 6 + 3];
    const int pr = cfg[j * 6 + 4], pc = cfg[j * 6 + 5];
    const int Hout = HW + 2 * pr - dr * (kr - 1);
    const int Wout = HW + 2 * pc - dc * (kc - 1);
    const int P = Hout * Wout;
    const float bj = b2[j];
    float w[16];
#pragma unroll
    for (int t = 0; t < 16; ++t) w[t] = w2[j * 16 + t];

    float s = 0.0f;
    for (int p = lane; p < P; p += 32) {
      const int r  = p / Wout;
      const int c0 = p - r * Wout;
      float acc = bj;
      for (int u = 0; u < kr; ++u) {
        const int rr = r + u * dr - pr;
        if ((unsigned)rr < (unsigned)HW) {
          const float* row = plane + rr * HW;
          for (int v = 0; v < kc; ++v) {
            const int cc = c0 + v * dc - pc;
            if ((unsigned)cc < (unsigned)HW)
              acc = fmaf(w[u * 4 + v], row[cc], acc);
          }
        }
      }
      s += fminf(fmaxf(acc, -PPV_T), PPV_T) * inv2t + 0.5f;
    }
    // wave32 reduction
    for (int off = 16; off > 0; off >>= 1)
      s += __shfl_xor(s, off, 32);
    if (lane == 0)
      out[bb * (N1 * N2) + i * N2 + j] = s / (float)P;
  }
}

extern "C" void kernel_launch(void* const* d_in, const int* in_sizes, int n_in,
                              void* d_out, int out_size, void* d_ws, size_t ws_size,
                              hipStream_t stream) {
  const float* x   = (const float*)d_in[0];   // (512,1,28,28)
  const float* w1  = (const float*)d_in[1];   // (10,1,3,3)
  const float* b1  = (const float*)d_in[2];   // (10,)
  const float* w2  = (const float*)d_in[3];   // (200,4,4)
  const float* b2  = (const float*)d_in[4];   // (200,)
  const int*   cfg = (const int*)d_in[5];     // (200,6) int32
  float* out   = (float*)d_out;               // (512, 2000)
  float* inter = (float*)d_ws;                // 10*512*784 f32 = 16.06 MB

  // Stage 1: one block per image, 8 waves sweep the 49 pixel tiles.
  rocket_layer1_wmma<<<BATCH, 256, 0, stream>>>(x, w1, b1, inter);
  // Stage 2: one block per (channel, image) plane.
  rocket_layer2_ppv<<<N1 * BATCH, 256, 0, stream>>>(inter, w2, b2, cfg, out);
}